// ConstraintLoss_53369263620137
// MI455X (gfx1250) — compile-verified
//
#include <hip/hip_runtime.h>
#include <hip/hip_bf16.h>

#define N_INT 40
#define DTC 0.25f
#define ROWS_PER_BLOCK 64
#define ROW_LEN 240   // 4*40 state + 2*40 control

typedef unsigned int u32x4 __attribute__((ext_vector_type(4)));
typedef int          i32x8 __attribute__((ext_vector_type(8)));
typedef int          i32x4 __attribute__((ext_vector_type(4)));
typedef float        f32x2 __attribute__((ext_vector_type(2)));
typedef float        f32x8 __attribute__((ext_vector_type(8)));

__global__ void cl_init_out(float* out) { out[0] = 0.0f; }

__global__ __launch_bounds__(ROWS_PER_BLOCK)
void cl_loss_kernel(const float* __restrict__ pred,
                    const float* __restrict__ targ,
                    const float* __restrict__ inp,
                    float* __restrict__ out,
                    int Brows, float inv_mse, float inv_b)
{
    __shared__ __align__(16) float sh[ROWS_PER_BLOCK * ROW_LEN];

    const int t = threadIdx.x;
    const long long row0 = (long long)blockIdx.x * ROWS_PER_BLOCK;

    // ---- TDM: DMA this block's 64 pred rows (one contiguous 61440B span) into LDS ----
    // Issued by wave 0 (EXEC is ignored for tensor ops; a duplicate issue by an
    // EXEC=0 wave would rewrite identical data, which is benign).
    {
        long long valid_rows = (long long)Brows - row0;
        if (valid_rows > ROWS_PER_BLOCK) valid_rows = ROWS_PER_BLOCK;
        if (valid_rows < 0) valid_rows = 0;
        const unsigned tile_elems  = ROWS_PER_BLOCK * ROW_LEN;           // 15360
        const unsigned valid_elems = (unsigned)(valid_rows * ROW_LEN);   // OOB -> zero fill

        if (t < 32) {
            unsigned lds_base = (unsigned)(uintptr_t)(&sh[0]); // low 32 bits of flat LDS addr = LDS offset
            unsigned long long ga = (unsigned long long)(uintptr_t)(pred + row0 * ROW_LEN);

            u32x4 g0;
            g0[0] = 1u;                                            // count=1, user descriptor
            g0[1] = lds_base;                                      // lds_addr
            g0[2] = (unsigned)(ga & 0xFFFFFFFFu);                  // global_addr[31:0]
            g0[3] = (unsigned)((ga >> 32) & 0x1FFFFFFu) | (2u << 30); // global_addr[56:32] | type=2

            i32x8 g1;
            g1[0] = (int)(2u << 16);                               // data_size=2 (4 bytes)
            g1[1] = (int)((valid_elems & 0xFFFFu) << 16);          // tensor_dim0[15:0]
            g1[2] = (int)(((valid_elems >> 16) & 0xFFFFu)          // tensor_dim0[31:16]
                          | (1u << 16));                           // tensor_dim1 = 1
            g1[3] = (int)((tile_elems & 0xFFFFu) << 16);           // tile_dim0 = 15360
            g1[4] = 1;                                             // tile_dim1 = 1, tile_dim2 = 0
            g1[5] = (int)tile_elems;                               // tensor_dim0_stride[31:0]
            g1[6] = 0;                                             // stride hi / dim1_stride lo
            g1[7] = 0;

            i32x4 g2 = {0, 0, 0, 0};
            i32x4 g3 = {0, 0, 0, 0};
            i32x8 g4 = {0, 0, 0, 0, 0, 0, 0, 0};

            __builtin_amdgcn_tensor_load_to_lds(g0, g1, g2, g3, g4, 0);
            __builtin_amdgcn_s_wait_tensorcnt(0);
        }
    }
    __syncthreads();

    // ---- per-row structured loss ----
    long long row = row0 + t;
    const bool valid = (row < (long long)Brows);
    const long long rr = valid ? row : 0;

    const float* __restrict__ shrow = &sh[t * ROW_LEN];
    const float* __restrict__ irow  = inp + rr * 13;

    float p0 = irow[0], p1 = irow[1], p2 = irow[2], p3 = irow[3];
    float ox0 = irow[4],  oy0 = irow[5],  rw0 = irow[6]  + 2.0f;
    float ox1 = irow[7],  oy1 = irow[8],  rw1 = irow[9]  + 2.0f;
    float ox2 = irow[10], oy2 = irow[11], rw2 = irow[12] + 2.0f;
    const float rad0 = rw0 * rw0, rad1 = rw1 * rw1, rad2v = rw2 * rw2;

    float r0 = 0.f, r1 = 0.f, r2 = 0.f, r3 = 0.f, obst = 0.f;
    #pragma unroll 4
    for (int i = 0; i < N_INT; ++i) {
        float x0v = shrow[4 * i + 0];
        float x1v = shrow[4 * i + 1];
        float x2v = shrow[4 * i + 2];
        float x3v = shrow[4 * i + 3];
        float av  = shrow[160 + 2 * i + 0];
        float wv  = shrow[160 + 2 * i + 1];

        float sn, cs;
        sincosf(p2, &sn, &cs);

        r0 += x0v - fmaf(p3 * cs, DTC, p0);
        r1 += x1v - fmaf(p3 * sn, DTC, p1);
        r2 += x2v - fmaf(wv, DTC, p2);
        r3 += x3v - fmaf(av, DTC, p3);

        float dx, dy;
        dx = x0v - ox0; dy = x1v - oy0; obst += sqrtf(fmaf(dx, dx, dy * dy)) - rad0;
        dx = x0v - ox1; dy = x1v - oy1; obst += sqrtf(fmaf(dx, dx, dy * dy)) - rad1;
        dx = x0v - ox2; dy = x1v - oy2; obst += sqrtf(fmaf(dx, dx, dy * dy)) - rad2v;

        p0 = x0v; p1 = x1v; p2 = x2v; p3 = x3v;
    }
    const float dyn_err = sqrtf(fmaf(r0, r0, fmaf(r1, r1, fmaf(r2, r2, r3 * r3))));

    // ---- MSE over this row: pred from LDS (float4), targets streamed from global ----
    float acc = 0.f;
    const float4* __restrict__ tg4 = (const float4*)(targ + rr * ROW_LEN);
    const float4* __restrict__ pg4 = (const float4*)shrow;
    #pragma unroll 4
    for (int j = 0; j < ROW_LEN / 4; ++j) {
        float4 pv = pg4[j];
        float4 tv = tg4[j];
        float d0 = pv.x - tv.x, d1 = pv.y - tv.y, d2 = pv.z - tv.z, d3 = pv.w - tv.w;
        acc += fmaf(d0, d0, fmaf(d1, d1, fmaf(d2, d2, d3 * d3)));
    }

    float val = valid ? fmaf(acc, inv_mse, (dyn_err + obst) * inv_b) : 0.0f;

    // ---- 32-lane sum via WMMA ones-matrix trick (EXEC all-1s here, no divergence) ----
    // A[m][0]=val(lane m), A[m][2]=val(lane m+16), A[m][1]=A[m][3]=0; B = ones
    // => D[m][n] = val(m) + val(m+16) for every column n.
    f32x2 Am; Am[0] = val; Am[1] = 0.0f;
    f32x2 Bm; Bm[0] = 1.0f; Bm[1] = 1.0f;
    f32x8 Cm = {};
    f32x8 Dm = __builtin_amdgcn_wmma_f32_16x16x4_f32(
        /*neg_a=*/false, Am, /*neg_b=*/false, Bm,
        /*c_mod=*/(short)0, Cm, /*reuse_a=*/false, /*reuse_b=*/false);

    float half = Dm[0] + Dm[1] + Dm[2] + Dm[3] + Dm[4] + Dm[5] + Dm[6] + Dm[7];
    float tot = half + __shfl_xor(half, 16, 32);   // lanes<16 hold rows 0-7, lanes>=16 rows 8-15

    if ((t & 31) == 0) atomicAdd(out, tot);
}

extern "C" void kernel_launch(void* const* d_in, const int* in_sizes, int n_in,
                              void* d_out, int out_size, void* d_ws, size_t ws_size,
                              hipStream_t stream)
{
    (void)n_in; (void)out_size; (void)d_ws; (void)ws_size;
    const float* pred = (const float*)d_in[0];
    const float* targ = (const float*)d_in[1];
    const float* inp  = (const float*)d_in[2];
    float* out = (float*)d_out;

    const int Brows = in_sizes[2] / 13;
    const float inv_mse = 1.0f / ((float)Brows * (float)ROW_LEN);
    const float inv_b   = 1.0f / (float)Brows;

    cl_init_out<<<1, 1, 0, stream>>>(out);

    const int nblocks = (Brows + ROWS_PER_BLOCK - 1) / ROWS_PER_BLOCK;
    cl_loss_kernel<<<nblocks, ROWS_PER_BLOCK, 0, stream>>>(
        pred, targ, inp, out, Brows, inv_mse, inv_b);
}